// STBlock_26044681683466
// MI455X (gfx1250) — compile-verified
//
#include <hip/hip_runtime.h>
#include <math.h>

// Problem dims (compile-time constants from the reference)
#define B_   32
#define N_   512
#define C_   64
#define T_   24
#define K_   3
#define F_   64
#define CT_  (C_*T_)   /* 1536 */
#define EPS_ 1e-5f

typedef __attribute__((ext_vector_type(2))) float v2f;
typedef __attribute__((ext_vector_type(8))) float v8f;
typedef __attribute__((ext_vector_type(4))) unsigned int u32x4_;
typedef __attribute__((ext_vector_type(8))) int i32x8_;
typedef __attribute__((ext_vector_type(4))) int i32x4_;

__device__ __forceinline__ float sigmoidf_(float v) { return 1.0f / (1.0f + __expf(-v)); }

// --------------------------- CDNA5 data movers -----------------------------
// Wave-level Tensor Data Mover: 2-D tile (tile_w x tile_h elements of 4B)
// from a row-major matrix with row stride `row_stride_elems` into LDS at
// byte offset `lds_byte`, padding +1 DWORD after every (1<<pad_ival_code)*8
// bytes (pad_ival_code: 4 -> every 32 DWORDs, 5 -> every 64 DWORDs).
__device__ __forceinline__ void tdm_load_2d(
    unsigned lds_byte, const void* gaddr,
    unsigned tile_w_elems, unsigned tile_h, unsigned row_stride_elems,
    unsigned pad_ival_code)
{
    const unsigned long long ga = (unsigned long long)gaddr;
    u32x4_ g0;
    g0[0] = 1u;                                    // count=1 valid user D#
    g0[1] = lds_byte;                              // lds_addr
    g0[2] = (unsigned)ga;                          // global_addr[31:0]
    g0[3] = (unsigned)(ga >> 32) | (2u << 30);     // global_addr[56:32] | type=2 (image)
    i32x8_ g1;
    g1[0] = (int)((2u << 16) | (1u << 20) | (pad_ival_code << 22)); // data_size=4B, pad_en, ival, amount=1DW
    g1[1] = (int)((row_stride_elems & 0xFFFFu) << 16);              // tensor_dim0[15:0]
    g1[2] = (int)((row_stride_elems >> 16) | (tile_h << 16));       // tensor_dim0[31:16] | tensor_dim1[15:0]
    g1[3] = (int)(tile_w_elems << 16);                              // tensor_dim1[31:16]=0 | tile_dim0
    g1[4] = (int)tile_h;                                            // tile_dim1 | tile_dim2=0
    g1[5] = (int)row_stride_elems;                                  // tensor_dim0_stride[31:0]
    g1[6] = 0;                                                      // stride0[47:32] | stride1[15:0]
    g1[7] = 0;                                                      // stride1[47:16]
    const i32x4_ z4 = {0, 0, 0, 0};
    const i32x8_ z8 = {0, 0, 0, 0, 0, 0, 0, 0};
#if __has_builtin(__builtin_amdgcn_tensor_load_to_lds)
    // clang-23 / therock-10.0 form: (g0, g1, g2, g3, g4, cpol)
    __builtin_amdgcn_tensor_load_to_lds(g0, g1, z4, z4, z8, 0);
#else
    (void)z4; (void)z8;
    asm volatile("tensor_load_to_lds %0, %1" :: "s"(g0), "s"(g1) : "memory");
#endif
}

__device__ __forceinline__ void tdm_wait0()
{
#if __has_builtin(__builtin_amdgcn_s_wait_tensorcnt)
    __builtin_amdgcn_s_wait_tensorcnt(0);
#else
    asm volatile("s_wait_tensorcnt 0x0" ::: "memory");
#endif
}

// Per-lane async copy of 16B global -> LDS (ASYNCcnt-tracked, no VGPR data).
__device__ __forceinline__ void async_copy_b128(void* lds_dst, const void* gsrc)
{
    const unsigned l = (unsigned)(unsigned long long)lds_dst;   // low 32b of generic = LDS offset
    const unsigned long long g = (unsigned long long)gsrc;
    asm volatile("global_load_async_to_lds_b128 %0, %1, off" :: "v"(l), "v"(g) : "memory");
}

__device__ __forceinline__ void async_wait0()
{
#if __has_builtin(__builtin_amdgcn_s_wait_asynccnt)
    __builtin_amdgcn_s_wait_asynccnt(0);
#else
    asm volatile("s_wait_asynccnt 0x0" ::: "memory");
#endif
}

// ---------------------------------------------------------------------------
// Batched fp32 GEMM on the WMMA pipe: C[bz] (+)= A[bz] (MxK) * B[bz] (KxN).
// 256 threads = 8 waves; block tile 32(M) x 64(N); K staged in 32-wide LDS
// panels filled by the Tensor Data Mover; each wave owns one 16x16 WMMA tile
// and issues 8 x V_WMMA_F32_16X16X4_F32 per panel.
// ---------------------------------------------------------------------------
#define BM 32
#define BN 64
#define KT 32

__global__ __launch_bounds__(256) void k_wmma_gemm(
    const float* __restrict__ A, const float* __restrict__ Bm, float* __restrict__ C,
    int M, int N, int K, long long sA, long long sB, long long sC, int accumulate)
{
    __shared__ __align__(16) float As[BM][KT + 1];   // +1 DWORD pad == TDM pad_interval 32DW
    __shared__ __align__(16) float Bs[KT][BN + 1];   // +1 DWORD pad == TDM pad_interval 64DW

    const int bz = blockIdx.z;
    A  += (long long)bz * sA;
    Bm += (long long)bz * sB;
    C  += (long long)bz * sC;

    const int m0   = blockIdx.y * BM;
    const int n0   = blockIdx.x * BN;
    const int tid  = threadIdx.x;
    const int lane = tid & 31;
    const int wave = tid >> 5;
    const int wm   = (wave >> 2) << 4;  // 0 / 16
    const int wn   = (wave & 3) << 4;   // 0 / 16 / 32 / 48
    const int l15  = lane & 15;
    const int lhi  = lane >> 4;         // 0 for lanes 0-15, 1 for 16-31

    const unsigned ldsA = (unsigned)(unsigned long long)(const void*)&As[0][0];
    const unsigned ldsB = (unsigned)(unsigned long long)(const void*)&Bs[0][0];

    // C/D layout: VGPR j -> row (j + 8*lhi), col l15 within the 16x16 tile
    const int crow = m0 + wm + lhi * 8;
    const int ccol = n0 + wn + l15;

    v8f acc = {};
    if (accumulate) {
#pragma unroll
        for (int j = 0; j < 8; ++j)
            acc[j] = C[(long long)(crow + j) * N + ccol];
    }

    // per-thread coordinates used only for next-panel prefetch
    const int ar = tid >> 3;            // 0..31
    const int ac = (tid & 7) << 2;
    const int br = tid >> 3;
    const int bc = (tid & 7) << 3;

    for (int k0 = 0; k0 < K; k0 += KT) {
        if (tid < 32) {   // wave 0 programs the TDM, waits on TENSORcnt
            tdm_load_2d(ldsA, A + (long long)m0 * K + k0, KT, BM, (unsigned)K, 4);
            tdm_load_2d(ldsB, Bm + (long long)k0 * N + n0, BN, KT, (unsigned)N, 5);
            tdm_wait0();
        }
        if (k0 + KT < K) {  // global_prefetch_b8 for the next panels
            __builtin_prefetch(&A[(long long)(m0 + ar) * K + (k0 + KT + ac)], 0, 1);
            __builtin_prefetch(&Bm[(long long)(k0 + KT + br) * N + (n0 + bc)], 0, 1);
        }
        __syncthreads();

#pragma unroll
        for (int kk = 0; kk < KT; kk += 4) {
            // A frag 16x4: lanes 0-15 -> K=kk,kk+1 ; lanes 16-31 -> K=kk+2,kk+3
            v2f a, b;
            a[0] = As[wm + l15][kk + (lhi << 1) + 0];
            a[1] = As[wm + l15][kk + (lhi << 1) + 1];
            // B frag 4x16: rows striped across lane halves within each VGPR
            b[0] = Bs[kk + 0 + lhi][wn + l15];
            b[1] = Bs[kk + 2 + lhi][wn + l15];
            acc = __builtin_amdgcn_wmma_f32_16x16x4_f32(
                false, a, false, b, (short)0, acc, false, false);
        }
        __syncthreads();
    }

#pragma unroll
    for (int j = 0; j < 8; ++j)
        C[(long long)(crow + j) * N + ccol] = acc[j];
}

// ---------------------------------------------------------------------------
// Spatial attention pre: lhs[b,n,t] = (sum_t' x W1) @ W2 ; rhs[b,n,t] = x . W3
// ---------------------------------------------------------------------------
__global__ __launch_bounds__(128) void k_spatial_pre(
    const float* __restrict__ x, const float* __restrict__ W1,
    const float* __restrict__ W2, const float* __restrict__ W3,
    float* __restrict__ lhs, float* __restrict__ rhs)
{
    __shared__ __align__(16) float xr[CT_];
    __shared__ float tmpc[C_];
    const long long bn = blockIdx.x;
    const float* xp = x + bn * CT_;
    for (int i = threadIdx.x * 4; i < CT_; i += 128 * 4)
        async_copy_b128(&xr[i], &xp[i]);
    async_wait0();
    __syncthreads();
    if (threadIdx.x < C_) {
        float s = 0.f;
#pragma unroll
        for (int t = 0; t < T_; ++t) s += xr[threadIdx.x * T_ + t] * W1[t];
        tmpc[threadIdx.x] = s;
    } else if (threadIdx.x < C_ + T_) {
        const int t = threadIdx.x - C_;
        float s = 0.f;
#pragma unroll
        for (int c = 0; c < C_; ++c) s += xr[c * T_ + t] * W3[c];
        rhs[bn * T_ + t] = s;
    }
    __syncthreads();
    if (threadIdx.x < T_) {
        const int t = threadIdx.x;
        float s = 0.f;
#pragma unroll
        for (int c = 0; c < C_; ++c) s += tmpc[c] * W2[c * T_ + t];
        lhs[bn * T_ + t] = s;
    }
}

// P[b,m,n] = sigmoid(lhs[b,m,:].rhs[b,n,:] + bs[m,n])
__global__ __launch_bounds__(256) void k_product_sig(
    const float* __restrict__ lhs, const float* __restrict__ rhs,
    const float* __restrict__ bs, float* __restrict__ P)
{
    __shared__ float Ls[16][T_];
    __shared__ float Rs[16][T_];
    const int b = blockIdx.z;
    const int m0 = blockIdx.y * 16, n0 = blockIdx.x * 16;
    for (int i = threadIdx.x; i < 16 * T_; i += 256) {
        Ls[i / T_][i % T_] = lhs[((long long)b * N_ + m0 + i / T_) * T_ + i % T_];
        Rs[i / T_][i % T_] = rhs[((long long)b * N_ + n0 + i / T_) * T_ + i % T_];
    }
    __syncthreads();
    const int tm = threadIdx.x >> 4, tn = threadIdx.x & 15;
    float s = 0.f;
#pragma unroll
    for (int t = 0; t < T_; ++t) s += Ls[tm][t] * Rs[tn][t];
    s += bs[(m0 + tm) * N_ + (n0 + tn)];
    P[((long long)b * N_ + m0 + tm) * N_ + (n0 + tn)] = sigmoidf_(s);
}

// In-place row softmax, row length 512
__global__ __launch_bounds__(256) void k_softmax512(float* __restrict__ S)
{
    __shared__ float red[256];
    float* row = S + (long long)blockIdx.x * N_;
    float v0 = row[threadIdx.x], v1 = row[threadIdx.x + 256];
    red[threadIdx.x] = fmaxf(v0, v1);
    __syncthreads();
    for (int s = 128; s > 0; s >>= 1) {
        if (threadIdx.x < s) red[threadIdx.x] = fmaxf(red[threadIdx.x], red[threadIdx.x + s]);
        __syncthreads();
    }
    const float m = red[0];
    __syncthreads();
    v0 = __expf(v0 - m); v1 = __expf(v1 - m);
    red[threadIdx.x] = v0 + v1;
    __syncthreads();
    for (int s = 128; s > 0; s >>= 1) {
        if (threadIdx.x < s) red[threadIdx.x] += red[threadIdx.x + s];
        __syncthreads();
    }
    const float inv = 1.0f / red[0];
    row[threadIdx.x] = v0 * inv;
    row[threadIdx.x + 256] = v1 * inv;
}

// Y[b,n,f,t] = sum_c x_s[b,n,c,t] * Theta[k,c,f]
__global__ __launch_bounds__(256) void k_theta(
    const float* __restrict__ xs, const float* __restrict__ Theta, int k,
    float* __restrict__ Y)
{
    __shared__ __align__(16) float xr[CT_];
    __shared__ __align__(16) float Th[C_ * F_];
    const long long bn = blockIdx.x;
    const float* xp = xs + bn * CT_;
    const float* Tp = Theta + (long long)k * C_ * F_;
    for (int i = threadIdx.x * 4; i < CT_; i += 256 * 4)
        async_copy_b128(&xr[i], &xp[i]);
    for (int i = threadIdx.x * 4; i < C_ * F_; i += 256 * 4)
        async_copy_b128(&Th[i], &Tp[i]);
    async_wait0();
    __syncthreads();
    for (int o = threadIdx.x; o < F_ * T_; o += 256) {
        const int f = o / T_, t = o % T_;
        float s = 0.f;
#pragma unroll
        for (int c = 0; c < C_; ++c) s += xr[c * T_ + t] * Th[c * F_ + f];
        Y[bn * (F_ * T_) + o] = s;
    }
}

__global__ __launch_bounds__(256) void k_relu(float* __restrict__ p, long long n)
{
    const long long i = (long long)blockIdx.x * 256 + threadIdx.x;
    if (i < n) p[i] = fmaxf(p[i], 0.f);
}

// tlhs[b,t,f] = sum_n h[b,n,f,t] * U1[n]   (one wave32 per output)
__global__ __launch_bounds__(256) void k_tlhs(
    const float* __restrict__ h, const float* __restrict__ U1, float* __restrict__ tlhs)
{
    const int id = blockIdx.x * 8 + (threadIdx.x >> 5);
    const int lane = threadIdx.x & 31;
    const int b = id / (T_ * F_);
    const int rem = id % (T_ * F_);
    const int t = rem / F_, f = rem % F_;
    float s = 0.f;
    for (int n = lane; n < N_; n += 32)
        s += h[(((long long)b * N_ + n) * F_ + f) * T_ + t] * U1[n];
    for (int off = 16; off > 0; off >>= 1) s += __shfl_down(s, off, 32);
    if (lane == 0) tlhs[((long long)b * T_ + t) * F_ + f] = s;
}

// lhsU[b,t,n] = sum_f tlhs[b,t,f] * U2[f,n]
__global__ __launch_bounds__(256) void k_lhsU2(
    const float* __restrict__ tlhs, const float* __restrict__ U2, float* __restrict__ lhsU)
{
    __shared__ float lr[F_];
    const long long bt = blockIdx.x;
    if (threadIdx.x < F_) lr[threadIdx.x] = tlhs[bt * F_ + threadIdx.x];
    __syncthreads();
    for (int n = threadIdx.x; n < N_; n += 256) {
        float s = 0.f;
#pragma unroll
        for (int f = 0; f < F_; ++f) s += lr[f] * U2[f * N_ + n];
        lhsU[bt * N_ + n] = s;
    }
}

// rhst[b,n,t] = sum_f U3[f] * h[b,n,f,t]
__global__ __launch_bounds__(256) void k_trhs(
    const float* __restrict__ h, const float* __restrict__ U3, float* __restrict__ rhst)
{
    const long long i = (long long)blockIdx.x * 256 + threadIdx.x;
    const int t = (int)(i % T_);
    const long long bn = i / T_;
    float s = 0.f;
#pragma unroll
    for (int f = 0; f < F_; ++f) s += U3[f] * h[(bn * F_ + f) * T_ + t];
    rhst[i] = s;
}

// Pt[b,t,s] = sigmoid(sum_n lhsU[b,t,n]*rhst[b,n,s] + be[t,s])  (wave per output)
__global__ __launch_bounds__(256) void k_tprod(
    const float* __restrict__ lhsU, const float* __restrict__ rhst,
    const float* __restrict__ be, float* __restrict__ Pt)
{
    const int id = blockIdx.x * 8 + (threadIdx.x >> 5);
    const int lane = threadIdx.x & 31;
    const int b = id / (T_ * T_);
    const int rem = id % (T_ * T_);
    const int t = rem / T_, s2 = rem % T_;
    float acc = 0.f;
    for (int n = lane; n < N_; n += 32)
        acc += lhsU[((long long)b * T_ + t) * N_ + n] * rhst[((long long)b * N_ + n) * T_ + s2];
    for (int off = 16; off > 0; off >>= 1) acc += __shfl_down(acc, off, 32);
    if (lane == 0) Pt[(long long)id] = sigmoidf_(acc + be[t * T_ + s2]);
}

// E[b] = softmax_rows(Ve @ Pt[b])
__global__ __launch_bounds__(256) void k_E(
    const float* __restrict__ Pt, const float* __restrict__ Ve, float* __restrict__ E)
{
    __shared__ float Ms[T_ * T_];
    const int b = blockIdx.x;
    for (int i = threadIdx.x; i < T_ * T_; i += 256) {
        const int t = i / T_, r = i % T_;
        float s = 0.f;
#pragma unroll
        for (int q = 0; q < T_; ++q) s += Ve[t * T_ + q] * Pt[(long long)b * T_ * T_ + q * T_ + r];
        Ms[i] = s;
    }
    __syncthreads();
    if (threadIdx.x < T_) {
        const int t = threadIdx.x;
        float m = -1e30f;
        for (int r = 0; r < T_; ++r) m = fmaxf(m, Ms[t * T_ + r]);
        float sum = 0.f;
        for (int r = 0; r < T_; ++r) sum += __expf(Ms[t * T_ + r] - m);
        const float inv = 1.0f / sum;
        for (int r = 0; r < T_; ++r)
            E[(long long)b * T_ * T_ + t * T_ + r] = __expf(Ms[t * T_ + r] - m) * inv;
    }
}

// hE[b,n,f,s] = sum_t h[b,n,f,t] * E[b,t,s]
__global__ __launch_bounds__(256) void k_tmix(
    const float* __restrict__ h, const float* __restrict__ E, float* __restrict__ hE)
{
    __shared__ __align__(16) float hr[F_ * T_];
    __shared__ __align__(16) float Eb[T_ * T_];
    const long long bn = blockIdx.x;
    const int b = (int)(bn / N_);
    for (int i = threadIdx.x * 4; i < F_ * T_; i += 256 * 4)
        async_copy_b128(&hr[i], &h[bn * F_ * T_ + i]);
    for (int i = threadIdx.x * 4; i < T_ * T_; i += 256 * 4)
        async_copy_b128(&Eb[i], &E[(long long)b * T_ * T_ + i]);
    async_wait0();
    __syncthreads();
    for (int o = threadIdx.x; o < F_ * T_; o += 256) {
        const int f = o / T_, s2 = o % T_;
        float acc = 0.f;
#pragma unroll
        for (int t = 0; t < T_; ++t) acc += hr[f * T_ + t] * Eb[t * T_ + s2];
        hE[bn * F_ * T_ + o] = acc;
    }
}

// conv(1,3) over t + bias + residual(x) + LayerNorm over f -> out (B,N,F,T)
__global__ __launch_bounds__(256) void k_conv_res_ln(
    const float* __restrict__ hE, const float* __restrict__ x,
    const float* __restrict__ w, const float* __restrict__ cbv,
    const float* __restrict__ lw, const float* __restrict__ lb,
    float* __restrict__ out)
{
    __shared__ __align__(16) float hr[F_ * T_];
    __shared__ __align__(16) float xr[F_ * T_];
    __shared__ float tmp[F_ * T_];
    __shared__ float mu[T_], rs[T_];
    const long long bn = blockIdx.x;
    for (int i = threadIdx.x * 4; i < F_ * T_; i += 256 * 4) {
        async_copy_b128(&hr[i], &hE[bn * F_ * T_ + i]);
        async_copy_b128(&xr[i], &x[bn * F_ * T_ + i]);   // C == F, same layout
    }
    async_wait0();
    __syncthreads();
    for (int o = threadIdx.x; o < F_ * T_; o += 256) {
        const int f = o / T_, t = o % T_;
        float acc = cbv[f];
        for (int fi = 0; fi < F_; ++fi) {
            const float* wp = w + ((long long)f * F_ + fi) * 3;
            const float a0 = (t > 0)      ? hr[fi * T_ + t - 1] : 0.f;
            const float a1 =                hr[fi * T_ + t];
            const float a2 = (t < T_ - 1) ? hr[fi * T_ + t + 1] : 0.f;
            acc += wp[0] * a0 + wp[1] * a1 + wp[2] * a2;
        }
        tmp[o] = acc + xr[o];
    }
    __syncthreads();
    if (threadIdx.x < T_) {
        const int t = threadIdx.x;
        float m = 0.f;
        for (int f = 0; f < F_; ++f) m += tmp[f * T_ + t];
        m *= (1.0f / F_);
        float v = 0.f;
        for (int f = 0; f < F_; ++f) { const float d = tmp[f * T_ + t] - m; v += d * d; }
        v *= (1.0f / F_);
        mu[t] = m;
        rs[t] = rsqrtf(v + EPS_);
    }
    __syncthreads();
    for (int o = threadIdx.x; o < F_ * T_; o += 256) {
        const int f = o / T_, t = o % T_;
        out[bn * F_ * T_ + o] = (tmp[o] - mu[t]) * rs[t] * lw[f] + lb[f];
    }
}

// ---------------------------------------------------------------------------
extern "C" void kernel_launch(void* const* d_in, const int* in_sizes, int n_in,
                              void* d_out, int out_size, void* d_ws, size_t ws_size,
                              hipStream_t stream)
{
    (void)in_sizes; (void)n_in; (void)out_size; (void)ws_size;
    const float* x     = (const float*)d_in[0];
    const float* cheb  = (const float*)d_in[1];
    const float* W1    = (const float*)d_in[2];
    const float* W2    = (const float*)d_in[3];
    const float* W3    = (const float*)d_in[4];
    const float* bs    = (const float*)d_in[5];
    const float* Vs    = (const float*)d_in[6];
    const float* U1    = (const float*)d_in[7];
    const float* U2    = (const float*)d_in[8];
    const float* U3    = (const float*)d_in[9];
    const float* be    = (const float*)d_in[10];
    const float* Ve    = (const float*)d_in[11];
    const float* Theta = (const float*)d_in[12];
    const float* cw    = (const float*)d_in[13];
    const float* cbv   = (const float*)d_in[14];
    const float* lw    = (const float*)d_in[15];
    const float* lb    = (const float*)d_in[16];
    float* out = (float*)d_out;

    float* ws = (float*)d_ws;
    long long off = 0;
    auto alloc = [&](long long nf) { float* p = ws + off; off += nf; return p; };
    float* lhs  = alloc((long long)B_ * N_ * T_);
    float* rhs  = alloc((long long)B_ * N_ * T_);
    float* P    = alloc((long long)B_ * N_ * N_);
    float* S    = alloc((long long)B_ * N_ * N_);
    float* xs   = alloc((long long)B_ * N_ * CT_);
    float* Y    = alloc((long long)B_ * N_ * CT_);
    float* h    = alloc((long long)B_ * N_ * CT_);
    float* tlhs = alloc((long long)B_ * T_ * F_);
    float* lhsU = alloc((long long)B_ * T_ * N_);
    float* rhst = alloc((long long)B_ * N_ * T_);
    float* Pt   = alloc((long long)B_ * T_ * T_);
    float* E    = alloc((long long)B_ * T_ * T_);
    float* hE   = xs;   // x_s is dead after the Theta/cheb stage -> reuse

    // ---- spatial attention
    k_spatial_pre<<<B_ * N_, 128, 0, stream>>>(x, W1, W2, W3, lhs, rhs);
    k_product_sig<<<dim3(N_ / 16, N_ / 16, B_), 256, 0, stream>>>(lhs, rhs, bs, P);
    k_wmma_gemm<<<dim3(N_ / BN, N_ / BM, B_), 256, 0, stream>>>(
        Vs, P, S, N_, N_, N_, 0LL, (long long)N_ * N_, (long long)N_ * N_, 0);
    k_softmax512<<<B_ * N_, 256, 0, stream>>>(S);

    // ---- spatial mixing: x_s = S @ x
    k_wmma_gemm<<<dim3(CT_ / BN, N_ / BM, B_), 256, 0, stream>>>(
        S, x, xs, N_, CT_, N_, (long long)N_ * N_, (long long)N_ * CT_, (long long)N_ * CT_, 0);

    // ---- Chebyshev graph conv: h = sum_k T_k @ (x_s . Theta_k), then ReLU
    for (int k = 0; k < K_; ++k) {
        k_theta<<<B_ * N_, 256, 0, stream>>>(xs, Theta, k, Y);
        k_wmma_gemm<<<dim3(CT_ / BN, N_ / BM, B_), 256, 0, stream>>>(
            cheb + (long long)k * N_ * N_, Y, h, N_, CT_, N_,
            0LL, (long long)N_ * CT_, (long long)N_ * CT_, k > 0 ? 1 : 0);
    }
    k_relu<<<(int)(((long long)B_ * N_ * CT_) / 256), 256, 0, stream>>>(h, (long long)B_ * N_ * CT_);

    // ---- temporal attention
    k_tlhs<<<(B_ * T_ * F_) / 8, 256, 0, stream>>>(h, U1, tlhs);
    k_lhsU2<<<B_ * T_, 256, 0, stream>>>(tlhs, U2, lhsU);
    k_trhs<<<(B_ * N_ * T_) / 256, 256, 0, stream>>>(h, U3, rhst);
    k_tprod<<<(B_ * T_ * T_) / 8, 256, 0, stream>>>(lhsU, rhst, be, Pt);
    k_E<<<B_, 256, 0, stream>>>(Pt, Ve, E);

    // ---- temporal mixing + conv + residual + LN
    k_tmix<<<B_ * N_, 256, 0, stream>>>(h, E, hE);
    k_conv_res_ln<<<B_ * N_, 256, 0, stream>>>(hE, x, cw, cbv, lw, lb, out);
}